// RWKV7Block_4475355922527
// MI455X (gfx1250) — compile-verified
//
#include <hip/hip_runtime.h>
#include <hip/hip_bf16.h>
#include <math.h>
#include <stdint.h>

// ---------------------------------------------------------------------------
// RWKV-7 DPLR chunked scan for gfx1250 (MI455X).
// wave32 + V_WMMA_F32_16X16X4_F32 + async global->LDS (ASYNCcnt) double buffer.
// Grid: one workgroup per (b,h) = 64 blocks; 128 threads = 4 waves.
// Each wave owns a 16-wide V-column slice of the 64x64 fp32 state (in LDS).
// ---------------------------------------------------------------------------

typedef float v2f __attribute__((ext_vector_type(2)));
typedef float v8f __attribute__((ext_vector_type(8)));
typedef int   v4i __attribute__((vector_size(16)));   // matches builtin param type

#define BSZ  2
#define HN   32
#define SN   1024
#define DN   64
#define CN   16            // chunk length
#define NCH  (SN / CN)
#define ST   72            // LDS row stride (floats): 288B, 16B aligned, != 64 banks
#define TST  18            // 16x16 T-tile row stride
#define HD   (HN * DN)     // 2048

// ---- async global->LDS 16B copy (GLOBAL_LOAD_ASYNC_TO_LDS_B128) -----------
typedef __attribute__((address_space(1))) v4i gas_v4i;   // global
typedef __attribute__((address_space(3))) v4i las_v4i;   // LDS

__device__ __forceinline__ void async_b128(const float* gp, float* lp) {
#if __has_builtin(__builtin_amdgcn_global_load_async_to_lds_b128)
  __builtin_amdgcn_global_load_async_to_lds_b128(
      (gas_v4i*)(uintptr_t)gp,
      (las_v4i*)(unsigned)(uintptr_t)lp,   // flat->LDS: addr[31:0]
      0, 0);
#else
  unsigned l32 = (unsigned)(uintptr_t)lp;
  asm volatile("global_load_async_to_lds_b128 %0, %1, off"
               :: "v"(l32), "v"(gp) : "memory");
#endif
}
__device__ __forceinline__ void wait_async0() {
#if __has_builtin(__builtin_amdgcn_s_wait_asynccnt)
  __builtin_amdgcn_s_wait_asynccnt(0);
#else
  asm volatile("s_wait_asynccnt 0x0" ::: "memory");
#endif
}

// ---- WMMA f32 16x16x4 wrapper ---------------------------------------------
__device__ __forceinline__ v8f wmma4(v2f a, v2f b, v8f c) {
  return __builtin_amdgcn_wmma_f32_16x16x4_f32(
      /*neg_a=*/false, a, /*neg_b=*/false, b,
      /*c_mod=*/(short)0, c, /*reuse_a=*/false, /*reuse_b=*/false);
}

// ---- operand loaders following the documented wave32 VGPR layouts ---------
__device__ __forceinline__ v2f ld_A(const float* M, int stride, int s) {
  int lane = threadIdx.x & 31;
  int m  = lane & 15;
  int kb = 4 * s + ((lane < 16) ? 0 : 2);
  v2f a; a.x = M[m * stride + kb]; a.y = M[m * stride + kb + 1];
  return a;
}
__device__ __forceinline__ v2f ld_A_T_scaled(const float* M, int stride, int s,
                                             int m0, const float* sc) {
  int lane = threadIdx.x & 31;
  int m  = m0 + (lane & 15);
  int kb = 4 * s + ((lane < 16) ? 0 : 2);
  float e = sc[m];
  v2f a; a.x = M[kb * stride + m] * e; a.y = M[(kb + 1) * stride + m] * e;
  return a;
}
__device__ __forceinline__ v2f ld_B(const float* M, int stride, int s, int n0) {
  int lane = threadIdx.x & 31;
  int n  = n0 + (lane & 15);
  int kb = 4 * s + ((lane < 16) ? 0 : 2);
  v2f b; b.x = M[kb * stride + n]; b.y = M[(kb + 1) * stride + n];
  return b;
}
__device__ __forceinline__ v2f ld_B_T(const float* M, int stride, int s, int n0) {
  int lane = threadIdx.x & 31;
  int n  = n0 + (lane & 15);
  int kb = 4 * s + ((lane < 16) ? 0 : 2);
  v2f b; b.x = M[n * stride + kb]; b.y = M[n * stride + kb + 1];
  return b;
}
__device__ __forceinline__ void st_C(float* M, int stride, int n0, v8f c) {
  int lane = threadIdx.x & 31;
  int n  = n0 + (lane & 15);
  int mb = (lane < 16) ? 0 : 8;
#pragma unroll
  for (int r = 0; r < 8; ++r) M[(mb + r) * stride + n] = c[r];
}
__device__ __forceinline__ void st_C_tril(float* M, int stride, v8f c) {
  int lane = threadIdx.x & 31;
  int n  = lane & 15;
  int mb = (lane < 16) ? 0 : 8;
#pragma unroll
  for (int r = 0; r < 8; ++r) {
    int m = mb + r;
    M[m * stride + n] = (n <= m) ? c[r] : 0.0f;  // inclusive causal mask
  }
}
__device__ __forceinline__ v8f ld_C_scaled(const float* M, int stride, int m0,
                                           int n0, const float* sc) {
  int lane = threadIdx.x & 31;
  int n  = n0 + (lane & 15);
  int mb = (lane < 16) ? 0 : 8;
  v8f c;
#pragma unroll
  for (int r = 0; r < 8; ++r) {
    int m = m0 + mb + r;
    c[r] = M[m * stride + n] * sc[m];
  }
  return c;
}

__global__ __launch_bounds__(128)
void rwkv7_dplr_chunked(const float* __restrict__ qg, const float* __restrict__ kg,
                        const float* __restrict__ vg, const float* __restrict__ ag,
                        const float* __restrict__ bg, const float* __restrict__ gkg,
                        const float* __restrict__ rk, const float* __restrict__ gg,
                        const float* __restrict__ gns, const float* __restrict__ gnbias,
                        float* __restrict__ out) {
  // double-buffered chunk tiles [2][CN][ST]; szb doubles as (gk -> Z)
  __shared__ __align__(16) float sqb[2][CN * ST], skb[2][CN * ST], svb[2][CN * ST];
  __shared__ __align__(16) float sab[2][CN * ST], sbb[2][CN * ST], szb[2][CN * ST];
  __shared__ __align__(16) float so[CN * ST];
  __shared__ __align__(16) float T1[CN * TST], T2[CN * TST];
  __shared__ __align__(16) float Hst[DN * ST];           // state [K=64][V=64]
  __shared__ float egend[DN], svec[CN], mu[CN], rstd[CN];

  const int tid  = threadIdx.x;
  const int wv   = tid >> 5;            // wave id 0..3
  const int bhix = blockIdx.x;          // 0..63
  const int bb   = bhix / HN;
  const int hh   = bhix % HN;
  const size_t base = (size_t)bhix * SN * DN;
  const int n0 = wv * 16;               // this wave's V-column slice

  // ---- prologue: kick off async DMA of chunk 0, then zero the state ----
  {
    const size_t cb = base;
#pragma unroll
    for (int it = 0; it < 2; ++it) {
      int p = tid + it * 128;                 // 0..255: 16B packet id
      int t = p >> 4, d4 = (p & 15) * 4;
      size_t gi = cb + (size_t)p * 4;
      int lo = t * ST + d4;
      async_b128(qg  + gi, &sqb[0][lo]);
      async_b128(kg  + gi, &skb[0][lo]);
      async_b128(vg  + gi, &svb[0][lo]);
      async_b128(ag  + gi, &sab[0][lo]);
      async_b128(bg  + gi, &sbb[0][lo]);
      async_b128(gkg + gi, &szb[0][lo]);
    }
  }
  for (int i = tid; i < DN * ST; i += 128) Hst[i] = 0.0f;

  for (int ch = 0; ch < NCH; ++ch) {
    const int cur = ch & 1;
    float* cq = sqb[cur]; float* ck = skb[cur]; float* cv = svb[cur];
    float* ca = sab[cur]; float* cbt = sbb[cur]; float* cz = szb[cur];

    // ---- wait for this chunk's DMA; make all waves' LDS writes visible ----
    wait_async0();
    __syncthreads();

    // ---- overlap: issue async DMA for the NEXT chunk into the other buffer
    if (ch + 1 < NCH) {
      const int nx = cur ^ 1;
      const size_t nb = base + (size_t)(ch + 1) * CN * DN;
#pragma unroll
      for (int it = 0; it < 2; ++it) {
        int p = tid + it * 128;
        int t = p >> 4, d4 = (p & 15) * 4;
        size_t gi = nb + (size_t)p * 4;
        int lo = t * ST + d4;
        async_b128(qg  + gi, &sqb[nx][lo]);
        async_b128(kg  + gi, &skb[nx][lo]);
        async_b128(vg  + gi, &svb[nx][lo]);
        async_b128(ag  + gi, &sab[nx][lo]);
        async_b128(bg  + gi, &sbb[nx][lo]);
        async_b128(gkg + gi, &szb[nx][lo]);
      }
    }

    // ---- correction scalar s_t = sum_d q*k*r_k (needs ORIGINAL q,k) ----
    if (tid < CN) {
      float s = 0.f;
      for (int d = 0; d < DN; ++d)
        s += cq[tid * ST + d] * ck[tid * ST + d] * rk[hh * DN + d];
      svec[tid] = s;
    }
    __syncthreads();

    // ---- per-column decay transform: a^=a*e^{g<t}, {k,b}^=·e^{-g<=t}, q^=·e^{g<=t}
    if (tid < DN) {
      float gc = 0.f;
#pragma unroll
      for (int t = 0; t < CN; ++t) {
        float gp = gc;
        gc += cz[t * ST + tid];
        float ei = __expf(-gc);
        ca[t * ST + tid]  *= __expf(gp);
        ck[t * ST + tid]  *= ei;
        cbt[t * ST + tid] *= ei;
        cq[t * ST + tid]  *= __expf(gc);
      }
      egend[tid] = __expf(gc);
    }
    __syncthreads();

    // ---- WMMA phase 1: Z_inter = A^ @ H_in ; T1 = A^ K^T ; T2 = A^ B^T ----
    {
      v8f c = {0.f, 0.f, 0.f, 0.f, 0.f, 0.f, 0.f, 0.f};
#pragma unroll
      for (int s = 0; s < 16; ++s)
        c = wmma4(ld_A(ca, ST, s), ld_B(Hst, ST, s, n0), c);
      st_C(cz, ST, n0, c);                   // cz reused: now holds Z
    }
    if (wv == 0) {
      v8f c = {0.f, 0.f, 0.f, 0.f, 0.f, 0.f, 0.f, 0.f};
#pragma unroll
      for (int s = 0; s < 16; ++s)
        c = wmma4(ld_A(ca, ST, s), ld_B_T(ck, ST, s, 0), c);
      st_C(T1, TST, 0, c);
    } else if (wv == 1) {
      v8f c = {0.f, 0.f, 0.f, 0.f, 0.f, 0.f, 0.f, 0.f};
#pragma unroll
      for (int s = 0; s < 16; ++s)
        c = wmma4(ld_A(ca, ST, s), ld_B_T(cbt, ST, s, 0), c);
      st_C(T2, TST, 0, c);
    }
    __syncthreads();

    // ---- triangular z recurrence: z_t += sum_{j<t} T1[t][j]*v_j + T2[t][j]*z_j
    if (tid < DN) {
      for (int t = 1; t < CN; ++t) {
        float acc = cz[t * ST + tid];
        for (int j = 0; j < t; ++j)
          acc += T1[t * TST + j] * cv[j * ST + tid]
               + T2[t * TST + j] * cz[j * ST + tid];
        cz[t * ST + tid] = acc;
      }
    }
    __syncthreads();

    // ---- WMMA phase 2a: S1 = tril(Q^ K^T), S2 = tril(Q^ B^T) ----
    if (wv == 0) {
      v8f c = {0.f, 0.f, 0.f, 0.f, 0.f, 0.f, 0.f, 0.f};
#pragma unroll
      for (int s = 0; s < 16; ++s)
        c = wmma4(ld_A(cq, ST, s), ld_B_T(ck, ST, s, 0), c);
      st_C_tril(T1, TST, c);
    } else if (wv == 1) {
      v8f c = {0.f, 0.f, 0.f, 0.f, 0.f, 0.f, 0.f, 0.f};
#pragma unroll
      for (int s = 0; s < 16; ++s)
        c = wmma4(ld_A(cq, ST, s), ld_B_T(cbt, ST, s, 0), c);
      st_C_tril(T2, TST, c);
    }
    __syncthreads();

    // ---- WMMA phase 2b: O = Q^ @ H_in + S1 @ V + S2 @ Z ----
    {
      v8f c = {0.f, 0.f, 0.f, 0.f, 0.f, 0.f, 0.f, 0.f};
#pragma unroll
      for (int s = 0; s < 16; ++s)
        c = wmma4(ld_A(cq, ST, s), ld_B(Hst, ST, s, n0), c);
#pragma unroll
      for (int s = 0; s < 4; ++s)
        c = wmma4(ld_A(T1, TST, s), ld_B(cv, ST, s, n0), c);
#pragma unroll
      for (int s = 0; s < 4; ++s)
        c = wmma4(ld_A(T2, TST, s), ld_B(cz, ST, s, n0), c);
      st_C(so, ST, n0, c);
    }
    __syncthreads();

    // ---- WMMA phase 3: H = diag(e^{g_end}) H + K~^T V + B~^T Z ----
#pragma unroll
    for (int mt = 0; mt < 4; ++mt) {
      v8f c = ld_C_scaled(Hst, ST, mt * 16, n0, egend);
#pragma unroll
      for (int s = 0; s < 4; ++s)
        c = wmma4(ld_A_T_scaled(ck, ST, s, mt * 16, egend), ld_B(cv, ST, s, n0), c);
#pragma unroll
      for (int s = 0; s < 4; ++s)
        c = wmma4(ld_A_T_scaled(cbt, ST, s, mt * 16, egend), ld_B(cz, ST, s, n0), c);
      st_C(Hst + mt * 16 * ST, ST, n0, c);
    }

    // ---- epilogue: per-head GroupNorm + correction + gate ----
    if (tid < CN) {
      float m1 = 0.f, m2 = 0.f;
      for (int d = 0; d < DN; ++d) {
        float x = so[tid * ST + d];
        m1 += x; m2 += x * x;
      }
      m1 *= (1.0f / DN);
      float var = m2 * (1.0f / DN) - m1 * m1;
      mu[tid] = m1;
      rstd[tid] = rsqrtf(var + 1e-5f);
    }
    __syncthreads();

#pragma unroll
    for (int it = 0; it < 8; ++it) {
      int idx = tid + it * 128;
      int t = idx >> 6, d = idx & 63;
      int col = hh * DN + d;
      float val = (so[t * ST + d] - mu[t]) * rstd[t] * gns[col] + gnbias[col]
                + svec[t] * cv[t * ST + d];
      size_t off = ((size_t)bb * SN + (size_t)ch * CN + t) * (size_t)HD + col;
      out[off] = val * gg[off];
    }
    __syncthreads();   // buffers reused next chunk
  }
}

extern "C" void kernel_launch(void* const* d_in, const int* in_sizes, int n_in,
                              void* d_out, int out_size, void* d_ws, size_t ws_size,
                              hipStream_t stream) {
  (void)in_sizes; (void)n_in; (void)d_ws; (void)ws_size; (void)out_size;
  const float* q   = (const float*)d_in[0];
  const float* k   = (const float*)d_in[1];
  const float* v   = (const float*)d_in[2];
  const float* a   = (const float*)d_in[3];
  const float* b   = (const float*)d_in[4];
  const float* gk  = (const float*)d_in[5];
  const float* r_k = (const float*)d_in[6];
  const float* g   = (const float*)d_in[7];
  const float* gns = (const float*)d_in[8];
  const float* gnb = (const float*)d_in[9];
  float* out = (float*)d_out;

  rwkv7_dplr_chunked<<<dim3(BSZ * HN), dim3(128), 0, stream>>>(
      q, k, v, a, b, gk, r_k, g, gns, gnb, out);
}